// ParameterizedKnowledgeSelector_24558622998611
// MI455X (gfx1250) — compile-verified
//
#include <hip/hip_runtime.h>
#include <cmath>

typedef __attribute__((ext_vector_type(2))) float v2f;
typedef __attribute__((ext_vector_type(8))) float v8f;
typedef __attribute__((__vector_size__(16))) int b128_t;   // matches async builtin param type

#define DDIM   1024
#define KROWS  512
#define BATCH  64
#define MTOT   (BATCH * KROWS)   // 32768 rows of the big GEMM
#define MTILE  128               // rows per workgroup (8 waves x 16 rows)
#define NBLK   256               // N columns processed per block-iteration
#define KCH    32                // K-reduction chunk staged through LDS
#define NKC    (DDIM / KCH)      // 32 K-chunks

#define APAD   36                // A row stride (words): bank-conflict-free, 16B aligned
#define BPAD   36                // B (transposed, n-major) row stride: same properties

// ---------------------------------------------------------------------------
// async global -> LDS helpers (CDNA5 ASYNCcnt path)
// ---------------------------------------------------------------------------
__device__ __forceinline__ void async_copy16(const float* g, float* l)
{
#if __has_builtin(__builtin_amdgcn_global_load_async_to_lds_b128)
    __builtin_amdgcn_global_load_async_to_lds_b128(
        (__attribute__((address_space(1))) b128_t*)g,
        (__attribute__((address_space(3))) b128_t*)l, 0, 0);
#else
    unsigned loff = (unsigned)(size_t)(__attribute__((address_space(3))) void*)l;
    asm volatile("global_load_async_to_lds_b128 %0, %1, off"
                 :: "v"(loff), "v"(g) : "memory");
#endif
}

template <int N>
__device__ __forceinline__ void wait_asynccnt()
{
#if __has_builtin(__builtin_amdgcn_s_wait_asynccnt)
    __builtin_amdgcn_s_wait_asynccnt(N);
#else
    asm volatile("s_wait_asynccnt %0" :: "n"(N) : "memory");
#endif
}

// ---------------------------------------------------------------------------
// kernelT: WzT[n][k] = Wz[k][n]   (one-time 1024x1024 tiled transpose, 8 MB)
// ---------------------------------------------------------------------------
__global__ __launch_bounds__(256) void transpose_kernel(
    const float* __restrict__ src, float* __restrict__ dst)
{
    __shared__ float tile[32][33];
    const int bx = blockIdx.x * 32;        // column base in src (n)
    const int by = blockIdx.y * 32;        // row base in src (k)
    const int tx = threadIdx.x & 31;
    const int ty = threadIdx.x >> 5;       // 8 rows per pass

    #pragma unroll
    for (int j = 0; j < 32; j += 8)
        tile[ty + j][tx] = src[(size_t)(by + ty + j) * DDIM + bx + tx];
    __syncthreads();
    #pragma unroll
    for (int j = 0; j < 32; j += 8)
        dst[(size_t)(bx + ty + j) * DDIM + by + tx] = tile[tx][ty + j];
}

// ---------------------------------------------------------------------------
// kernel0: c[b][e] = sum_d u[b][d] * Wu[d][e]   (64x1024 output, tiny GEMV)
// ---------------------------------------------------------------------------
__global__ __launch_bounds__(256) void u_gemv_kernel(
    const float* __restrict__ x, const float* __restrict__ Wu, float* __restrict__ c)
{
    __shared__ float ulds[8 * DDIM];               // 32 KB: 8 rows of u
    const int tid = threadIdx.x;
    const int b0  = blockIdx.x * 8;
    const int e0  = blockIdx.y * 64;

    for (int i = tid; i < 8 * DDIM; i += 256) {
        int bb = i >> 10, d = i & (DDIM - 1);
        ulds[i] = x[((size_t)(b0 + bb) * 513) * DDIM + d];   // u = x[:,0,:]
    }
    __syncthreads();

    const int e  = e0 + (tid & 63);
    const int bs = tid >> 6;                       // 0..3
    float acc0 = 0.f, acc1 = 0.f;
    for (int d = 0; d < DDIM; ++d) {
        float w = Wu[(size_t)d * DDIM + e];        // coalesced across the 64 e-lanes
        acc0 = fmaf(ulds[bs * DDIM + d],       w, acc0);
        acc1 = fmaf(ulds[(bs + 4) * DDIM + d], w, acc1);
    }
    c[(size_t)(b0 + bs)     * DDIM + e] = acc0;
    c[(size_t)(b0 + bs + 4) * DDIM + e] = acc1;
}

// ---------------------------------------------------------------------------
// kernel1: fused  q[b,k] = sum_n tanh( (z Wz)[row,n] + c[b,n] ) * bias[n]
//   V_WMMA_F32_16X16X4_F32; double-buffered async global->LDS staging;
//   B staged K-major (from WzT) so every fragment is one aligned ds_load_b64.
// ---------------------------------------------------------------------------
__device__ __forceinline__ void issue_stage(
    const float* __restrict__ x, const float* __restrict__ WzT,
    float* Abuf, float* Bbuf, int row0, int n0blk, int kbase, int tid)
{
    // A chunk: 128 rows x 32 K floats = 1024 x 16B items, 4 per thread
    #pragma unroll
    for (int it = 0; it < 4; ++it) {
        int i   = tid + it * 256;
        int row = i >> 3;
        int kk4 = (i & 7) << 2;
        int r   = row0 + row;
        int b   = r >> 9;
        int k   = r & (KROWS - 1);
        const float* src = x + ((size_t)b * 513 + (k + 1)) * DDIM + kbase + kk4;
        async_copy16(src, &Abuf[row * APAD + kk4]);
    }
    // B chunk (K-major): 256 n-rows x 32 K floats = 2048 x 16B items, 8 per thread
    #pragma unroll
    for (int it = 0; it < 8; ++it) {
        int i   = tid + it * 256;
        int nn  = i >> 3;
        int kk4 = (i & 7) << 2;
        const float* src = WzT + (size_t)(n0blk + nn) * DDIM + kbase + kk4;
        async_copy16(src, &Bbuf[nn * BPAD + kk4]);
    }
}

__global__ __launch_bounds__(256) void fused_gemm_tanh_dot_kernel(
    const float* __restrict__ x, const float* __restrict__ WzT,
    const float* __restrict__ cbuf, const float* __restrict__ bias,
    float* __restrict__ q)
{
    __shared__ float Alds[2][MTILE * APAD];        // 2 x 18 KB
    __shared__ float Blds[2][NBLK * BPAD];         // 2 x 36 KB (n-major, K-contig)

    const int tid  = threadIdx.x;
    const int wave = tid >> 5;                     // 0..7
    const int lane = tid & 31;
    const int l15  = lane & 15;
    const int hi   = lane >> 4;                    // 0: K=k0,k0+1  1: K=k0+2,k0+3
    const int row0 = blockIdx.x * MTILE;           // global GEMM row base

    float qacc[8] = {0.f, 0.f, 0.f, 0.f, 0.f, 0.f, 0.f, 0.f};

    const int tile_r = row0 + wave * 16;           // 16 rows owned by this wave
    const int bidx   = tile_r >> 9;                // batch index (constant in tile)

    for (int nb = 0; nb < DDIM / NBLK; ++nb) {
        const int n0blk = nb * NBLK;

        v8f acc[NBLK / 16];
        #pragma unroll
        for (int t = 0; t < NBLK / 16; ++t) {
            v8f z = {0.f, 0.f, 0.f, 0.f, 0.f, 0.f, 0.f, 0.f};
            acc[t] = z;
        }

        // prologue: fill stage 0
        issue_stage(x, WzT, Alds[0], Blds[0], row0, n0blk, 0, tid);

        for (int kc = 0; kc < NKC; ++kc) {
            const int cur = kc & 1;

            if (kc + 1 < NKC) {
                issue_stage(x, WzT, Alds[(kc + 1) & 1], Blds[(kc + 1) & 1],
                            row0, n0blk, (kc + 1) * KCH, tid);
                wait_asynccnt<12>();               // current stage done; next in flight
            } else {
                wait_asynccnt<0>();
            }
            __syncthreads();                       // all waves' fills visible

            const float* Abuf  = Alds[cur];
            const float* Bbuf  = Blds[cur];
            const float* abase = &Abuf[(wave * 16 + l15) * APAD + 2 * hi];
            const float* bbase = &Bbuf[l15 * BPAD + 2 * hi];

            #pragma unroll
            for (int ks = 0; ks < KCH / 4; ++ks) {
                const int k0 = ks * 4;
                v2f a = *(const v2f*)(abase + k0);                 // ds_load_b64
                #pragma unroll
                for (int nt = 0; nt < NBLK / 16; ++nt) {
                    v2f b = *(const v2f*)(bbase + nt * 16 * BPAD + k0); // ds_load_b64
                    acc[nt] = __builtin_amdgcn_wmma_f32_16x16x4_f32(
                        false, a, false, b, (short)0, acc[nt], false, false);
                }
            }
            __syncthreads();                       // buffer safe for reuse at kc+2
        }

        // Epilogue of this N-block: add u-term, tanh, fold into q partials.
        #pragma unroll
        for (int nt = 0; nt < NBLK / 16; ++nt) {
            const int n  = n0blk + nt * 16 + l15;
            const float cv = cbuf[(size_t)bidx * DDIM + n];
            const float bv = bias[n];
            #pragma unroll
            for (int j = 0; j < 8; ++j) {
                float s = tanhf(acc[nt][j] + cv);
                qacc[j] = fmaf(s, bv, qacc[j]);
            }
        }
    }

    // Reduce q partials across the 16-lane groups of the C layout.
    #pragma unroll
    for (int j = 0; j < 8; ++j) {
        float v = qacc[j];
        v += __shfl_xor(v, 1, 32);
        v += __shfl_xor(v, 2, 32);
        v += __shfl_xor(v, 4, 32);
        v += __shfl_xor(v, 8, 32);
        qacc[j] = v;
    }
    if (l15 == 0) {
        const int m = hi ? 8 : 0;                  // lane 0 -> rows 0..7, lane 16 -> rows 8..15
        const int k = tile_r & (KROWS - 1);
        #pragma unroll
        for (int j = 0; j < 8; ++j)
            q[(size_t)bidx * KROWS + k + m + j] = qacc[j];
    }
}

// ---------------------------------------------------------------------------
// kernel2: row-wise stable softmax over k=512 per batch row
// ---------------------------------------------------------------------------
__global__ __launch_bounds__(256) void softmax_kernel(
    const float* __restrict__ q, float* __restrict__ out)
{
    __shared__ float red[256];
    const int b   = blockIdx.x;
    const int tid = threadIdx.x;
    const float v0 = q[(size_t)b * KROWS + tid];
    const float v1 = q[(size_t)b * KROWS + 256 + tid];

    red[tid] = fmaxf(v0, v1);
    __syncthreads();
    for (int s = 128; s > 0; s >>= 1) {
        if (tid < s) red[tid] = fmaxf(red[tid], red[tid + s]);
        __syncthreads();
    }
    const float mx = red[0];
    __syncthreads();

    const float e0 = expf(v0 - mx);
    const float e1 = expf(v1 - mx);
    red[tid] = e0 + e1;
    __syncthreads();
    for (int s = 128; s > 0; s >>= 1) {
        if (tid < s) red[tid] += red[tid + s];
        __syncthreads();
    }
    const float inv = 1.0f / red[0];
    out[(size_t)b * KROWS + tid]       = e0 * inv;
    out[(size_t)b * KROWS + 256 + tid] = e1 * inv;
}

// ---------------------------------------------------------------------------
extern "C" void kernel_launch(void* const* d_in, const int* in_sizes, int n_in,
                              void* d_out, int out_size, void* d_ws, size_t ws_size,
                              hipStream_t stream)
{
    const float* x    = (const float*)d_in[0];              // (64, 513, 1024)
    const float* dw   = (const float*)d_in[1];              // (2048, 1024)
    const float* bias = (const float*)d_in[2];              // (1024, 1)
    const float* Wz   = dw;                                 // rows 0..1023
    const float* Wu   = dw + (size_t)DDIM * DDIM;           // rows 1024..2047

    float* c   = (float*)d_ws;                              // 64*1024   f32 u-term
    float* q   = c + (size_t)BATCH * DDIM;                  // 64*512    f32 logits
    float* WzT = q + (size_t)BATCH * KROWS;                 // 1024*1024 f32 Wz^T

    dim3 gT(DDIM / 32, DDIM / 32);
    transpose_kernel<<<gT, 256, 0, stream>>>(Wz, WzT);

    dim3 g0(BATCH / 8, DDIM / 64);
    u_gemv_kernel<<<g0, 256, 0, stream>>>(x, Wu, c);

    fused_gemm_tanh_dot_kernel<<<MTOT / MTILE, 256, 0, stream>>>(x, WzT, c, bias, q);
    softmax_kernel<<<BATCH, 256, 0, stream>>>(q, (float*)d_out);
}